// Linearv2_81286551044362
// MI455X (gfx1250) — compile-verified
//
#include <hip/hip_runtime.h>

// CDNA5 / gfx1250, wave32. Uses v_wmma_f32_16x16x4_f32.
typedef __attribute__((ext_vector_type(2))) float v2f;
typedef __attribute__((ext_vector_type(8))) float v8f;

#define MUL     4096
#define KSPLIT  8                 // waves per block; each covers 4096/8 = 512 of K
#define KCHUNK  (MUL / KSPLIT)

// Grid: 4 paths * (4096/32) v-tile-pairs = 512 blocks of 256 threads.
// Each wave: two 16-wide v tiles (acc0: v0..v0+15, acc1: v0+16..v0+31),
// K-chunk of 512 -> 128 iterations of k-step 4 -> 2 WMMAs per iteration.
__global__ __launch_bounds__(256)
void irreps_linear_wmma(const float* __restrict__ x,
                        const float* __restrict__ w,
                        float* __restrict__ out) {
    __shared__ float red[KSPLIT * 512];   // 16 KB: 8 waves * 2 tiles * 8 vgprs * 32 lanes

    const int block = blockIdx.x;
    const int path  = block >> 7;          // 128 blocks per path
    const int pair  = block & 127;
    const int v0    = pair << 5;           // 32 output rows (v) per block

    const int       d      = 2 * path + 1;                 // 1,3,5,7
    const int       io_off = (path * path) << 12;          // 4096 * path^2 (in == out layout)
    const long long w_off  = (long long)path << 24;        // 4096*4096 weights per path

    const int tid  = threadIdx.x;
    const int wave = tid >> 5;
    const int lane = tid & 31;
    const int half = lane >> 4;            // K sub-half per WMMA layout
    const int r    = lane & 15;            // M (for A) / N (for B) within tile

    // Masked/clamped X column so control flow stays uniform (EXEC all-ones for WMMA).
    const int   i_col = (r < d) ? r : (d - 1);
    const float maskf = (r < d) ? 1.0f : 0.0f;

    // Per-lane streaming pointers for this wave's K-chunk.
    const int u0 = wave * KCHUNK + 2 * half;               // first K row this lane touches
    const float* wp = w + w_off + (long long)u0 * MUL + v0 + r;   // A-frag source
    const float* xp = x + io_off + u0 * d + i_col;                // B-frag source

    v8f acc0 = {};
    v8f acc1 = {};

    #pragma unroll 4
    for (int it = 0; it < KCHUNK / 4; ++it) {
        v2f a0, a1, b;
        // A tile 0: W[u+2h, v0+r], W[u+2h+1, v0+r]
        a0.x = wp[0];
        a0.y = wp[MUL];
        // A tile 1: columns v0+16..v0+31 (same cacheline as tile 0)
        a1.x = wp[16];
        a1.y = wp[MUL + 16];
        // B: X[u+2h, n], X[u+2h+1, n], zeroed for n >= d
        b.x = xp[0] * maskf;
        b.y = xp[d] * maskf;

        acc0 = __builtin_amdgcn_wmma_f32_16x16x4_f32(false, a0, false, b,
                                                     (short)0, acc0, false, false);
        acc1 = __builtin_amdgcn_wmma_f32_16x16x4_f32(false, a1, false, b,
                                                     (short)0, acc1, false, false);

        wp += 4 * MUL;   // next k-step: 4 rows down
        xp += 4 * d;
    }

    // Dump per-wave partials to LDS: red[wave*512 + tile*256 + j*32 + lane]
    {
        float* my = &red[wave * 512 + lane];
        #pragma unroll
        for (int j = 0; j < 8; ++j) {
            my[j * 32]       = acc0[j];
            my[256 + j * 32] = acc1[j];
        }
    }
    __syncthreads();

    // Deterministic fixed-order reduction across the 8 K-split waves + store.
    const float norm = 0.015625f;          // sqrt(1/4096) exactly
    for (int s = tid; s < 512; s += 256) {
        float sum = 0.0f;
        #pragma unroll
        for (int wv = 0; wv < KSPLIT; ++wv)
            sum += red[wv * 512 + s];
        const int tile = s >> 8;
        const int j    = (s >> 5) & 7;
        const int ln   = s & 31;
        const int M    = j + ((ln >> 4) << 3);   // C/D layout: VGPR j holds M=j / M=j+8
        const int N    = ln & 15;
        if (N < d)
            out[io_off + (v0 + tile * 16 + M) * d + N] = sum * norm;
    }
}

extern "C" void kernel_launch(void* const* d_in, const int* in_sizes, int n_in,
                              void* d_out, int out_size, void* d_ws, size_t ws_size,
                              hipStream_t stream) {
    (void)in_sizes; (void)n_in; (void)d_ws; (void)ws_size; (void)out_size;
    const float* x   = (const float*)d_in[0];   // "input":   65536 f32
    const float* wts = (const float*)d_in[1];   // "weights": 67108864 f32
    float* out = (float*)d_out;                 // 65536 f32

    dim3 grid(512);   // 4 paths * 128 v-tile-pairs
    dim3 block(256);  // 8 wave32
    irreps_linear_wmma<<<grid, block, 0, stream>>>(x, wts, out);
}